// Decoder_49151605735822
// MI455X (gfx1250) — compile-verified
//
#include <hip/hip_runtime.h>
#include <hip/hip_bf16.h>
#include <math.h>

#define B_  32
#define S_  4096
#define D_  1024
#define EPSF 1e-5f

typedef float v2f __attribute__((ext_vector_type(2)));
typedef float v8f __attribute__((ext_vector_type(8)));

__device__ __forceinline__ float gelu_exact(float x) {
    // 0.5 * x * (1 + erf(x / sqrt(2)))
    return 0.5f * x * (1.0f + erff(x * 0.70710678118654752440f));
}

// ---------------------------------------------------------------------------
// Kernel 0: zero the LN accumulators (sum1[32], sq1[32], sum2[32], sq2[32])
// ---------------------------------------------------------------------------
__global__ void zero_acc_kernel(float* __restrict__ acc) {
    acc[threadIdx.x] = 0.0f;
}

// ---------------------------------------------------------------------------
// Kernel 1: base1[s,k] = b1[k] + sum_d be1[s,d]*W1[d,k] - m1*r1*sum_d g1[s,d]*W1[d,k]
// Batch-independent part of (LN1(one_hot) @ W1 + b1).
// 4 s-rows per block, 128 k-threads; g1/be1 rows staged in LDS (broadcast reads).
// ---------------------------------------------------------------------------
__global__ __launch_bounds__(128) void precompute_base1_kernel(
    const float* __restrict__ g1, const float* __restrict__ be1,
    const float* __restrict__ W1, const float* __restrict__ b1,
    float* __restrict__ base1)
{
    __shared__ float sg[4 * D_];
    __shared__ float sb[4 * D_];
    const int tid = threadIdx.x;
    const int s0  = blockIdx.x * 4;

    for (int i = tid; i < 4 * D_; i += 128) {
        sg[i] = g1[(size_t)s0 * D_ + i];
        sb[i] = be1[(size_t)s0 * D_ + i];
    }
    __syncthreads();

    float ag[4] = {0.f, 0.f, 0.f, 0.f};
    float ab[4] = {0.f, 0.f, 0.f, 0.f};
    for (int d = 0; d < D_; ++d) {
        const float w = W1[d * 128 + tid];          // coalesced, L2-resident
        #pragma unroll
        for (int r = 0; r < 4; ++r) {
            ag[r] = fmaf(sg[r * D_ + d], w, ag[r]);
            ab[r] = fmaf(sb[r * D_ + d], w, ab[r]);
        }
    }
    const float m1 = 1.0f / (float)D_;
    const float r1 = rsqrtf(m1 * (1.0f - m1) + EPSF);
    const float bk = b1[tid];
    #pragma unroll
    for (int r = 0; r < 4; ++r)
        base1[(size_t)(s0 + r) * 128 + tid] = bk + ab[r] - m1 * r1 * ag[r];
}

// ---------------------------------------------------------------------------
// Kernel 2: stage 1. One block per (b,s) row:
//   h[b,s,k] = gelu( r1*g1[s,idx]*W1[idx,k] + base1[s,k] )
// Also accumulates per-batch sum / sumsq for LN2.
// ---------------------------------------------------------------------------
__global__ __launch_bounds__(128) void stage1_kernel(
    const int* __restrict__ inputs, const float* __restrict__ g1,
    const float* __restrict__ W1, const float* __restrict__ base1,
    float* __restrict__ hbuf, float* __restrict__ acc)
{
    __shared__ float red[256];
    const int tid = threadIdx.x;
    const int bs  = blockIdx.x;
    const int b   = bs >> 12;           // S_ == 4096
    const int s   = bs & (S_ - 1);

    const float m1 = 1.0f / (float)D_;
    const float r1 = rsqrtf(m1 * (1.0f - m1) + EPSF);

    const int   idx   = inputs[bs];                          // scalar broadcast
    const float scale = r1 * g1[(size_t)s * D_ + idx];       // scalar broadcast
    const float y = fmaf(scale, W1[idx * 128 + tid], base1[(size_t)s * 128 + tid]);
    const float h = gelu_exact(y);
    hbuf[(size_t)bs * 128 + tid] = h;

    red[tid] = h; red[128 + tid] = h * h;
    __syncthreads();
    for (int off = 64; off > 0; off >>= 1) {
        if (tid < off) { red[tid] += red[tid + off]; red[128 + tid] += red[128 + tid + off]; }
        __syncthreads();
    }
    if (tid == 0) {
        atomicAdd(&acc[b],      red[0]);
        atomicAdd(&acc[32 + b], red[128]);
    }
}

// ---------------------------------------------------------------------------
// Kernel 3: stage 2 (the WMMA kernel).
//   hn = LN2(h) * g2 + be2 ;  z = gelu(hn @ W2 + b2)   [K=128 -> N=64], fp32.
// 64 threads = 2 waves per block; each wave owns a 16-row s-tile.
// GEMM done with V_WMMA_F32_16X16X4_F32 (fragments built from LDS per the
// ISA f32 layouts; LDS row stride 132 floats -> conflict-free b64 reads).
// Accumulates per-batch sum / sumsq of gelu output for LN3.
// ---------------------------------------------------------------------------
__global__ __launch_bounds__(64) void stage2_wmma_kernel(
    const float* __restrict__ hbuf, const float* __restrict__ g2,
    const float* __restrict__ be2, const float* __restrict__ W2,
    const float* __restrict__ b2, float* __restrict__ zbuf,
    float* __restrict__ acc)
{
    extern __shared__ float sm[];
    float* w2T = sm;                                  // 64 rows(n) x 132 (k, padded)
    const int tid  = threadIdx.x;
    const int wave = tid >> 5;
    const int lane = tid & 31;
    const int half = lane >> 4;
    const int lr   = lane & 15;
    float* hnT = sm + 64 * 132 + wave * (16 * 132);   // 16 rows(s) x 132 (k, padded)
    float* red = sm + 64 * 132 + 2 * (16 * 132);      // 128 floats

    const int b  = blockIdx.y;
    const int s0 = blockIdx.x * 32 + wave * 16;

    // LN2 scalars (per batch, over S x 128)
    const float invN = 1.0f / (float)((size_t)S_ * 128);
    const float mean = acc[b] * invN;
    const float var  = acc[32 + b] * invN - mean * mean;
    const float rstd = rsqrtf(var + EPSF);

    // Stage W2 transposed into LDS: w2T[n*132 + k] = W2[k*64 + n]
    for (int i = tid; i < 128 * 64; i += 64) {
        const int k = i >> 6, n = i & 63;
        w2T[n * 132 + k] = W2[i];
    }

    // Load + normalize the 16x128 activation tile (float4, fully coalesced)
    for (int r = 0; r < 16; ++r) {
        const size_t row = (size_t)b * S_ + (size_t)(s0 + r);
        const float4 h4 = *(const float4*)(hbuf + row * 128 + lane * 4);
        const float4 gg = *(const float4*)(g2  + (size_t)(s0 + r) * 128 + lane * 4);
        const float4 bb = *(const float4*)(be2 + (size_t)(s0 + r) * 128 + lane * 4);
        float* dst = hnT + r * 132 + lane * 4;
        dst[0] = (h4.x - mean) * rstd * gg.x + bb.x;
        dst[1] = (h4.y - mean) * rstd * gg.y + bb.y;
        dst[2] = (h4.z - mean) * rstd * gg.z + bb.z;
        dst[3] = (h4.w - mean) * rstd * gg.w + bb.w;
    }
    __syncthreads();

    // WMMA: A lane(<16) holds {A[M=lr,k], A[M=lr,k+1]}, lane(>=16) K+2/K+3;
    //       B mirrored; D VGPR v holds row M = v + 8*half, col N = lr.
    float lsum = 0.f, lsq = 0.f;
    const float* aRow = hnT + lr * 132 + 2 * half;
    #pragma unroll
    for (int nt = 0; nt < 4; ++nt) {
        const int n0 = nt * 16;
        const float* bRow = w2T + (n0 + lr) * 132 + 2 * half;
        v8f accv = {0.f, 0.f, 0.f, 0.f, 0.f, 0.f, 0.f, 0.f};
        #pragma unroll
        for (int t = 0; t < 32; ++t) {                 // K = 128 in steps of 4
            const v2f a = *(const v2f*)(aRow + 4 * t);
            const v2f w = *(const v2f*)(bRow + 4 * t);
            accv = __builtin_amdgcn_wmma_f32_16x16x4_f32(
                false, a, false, w, (short)0, accv, false, false);
        }
        const float bias = b2[n0 + lr];
        #pragma unroll
        for (int v = 0; v < 8; ++v) {
            const int srow = s0 + v + 8 * half;
            const float z = gelu_exact(accv[v] + bias);
            zbuf[((size_t)b * S_ + srow) * 64 + n0 + lr] = z;
            lsum += z; lsq += z * z;
        }
    }

    red[tid] = lsum; red[64 + tid] = lsq;
    __syncthreads();
    if (tid == 0) {
        float s = 0.f, q = 0.f;
        for (int i = 0; i < 64; ++i) { s += red[i]; q += red[64 + i]; }
        atomicAdd(&acc[64 + b], s);
        atomicAdd(&acc[96 + b], q);
    }
}

// ---------------------------------------------------------------------------
// Kernel 4: stage 3.  out[b,o,s] = LN3(z)*g3+be3 @ W3 + b3, transposed store.
// z tile staged through LDS (padded stride 68) for coalesced global reads.
// ---------------------------------------------------------------------------
__global__ __launch_bounds__(128) void stage3_kernel(
    const float* __restrict__ zbuf, const float* __restrict__ g3,
    const float* __restrict__ be3, const float* __restrict__ W3,
    const float* __restrict__ b3, const float* __restrict__ acc,
    float* __restrict__ out)
{
    __shared__ float zt[128 * 68];
    const int tid   = threadIdx.x;
    const int b     = blockIdx.y;
    const int sBase = blockIdx.x * 128;

    const float invN = 1.0f / (float)((size_t)S_ * 64);
    const float mean = acc[64 + b] * invN;
    const float var  = acc[96 + b] * invN - mean * mean;
    const float rstd = rsqrtf(var + EPSF);

    for (int i = 0; i < 16; ++i) {
        const int row = i * 8 + (tid >> 4);
        const int c   = (tid & 15) * 4;
        const float4 z4 = *(const float4*)(zbuf + ((size_t)b * S_ + sBase + row) * 64 + c);
        float* dst = zt + row * 68 + c;
        dst[0] = z4.x; dst[1] = z4.y; dst[2] = z4.z; dst[3] = z4.w;
    }
    __syncthreads();

    const int s = sBase + tid;
    const float* zr = zt + tid * 68;
    float o0 = b3[0], o1 = b3[1];
    for (int m = 0; m < 64; ++m) {
        const float hn = (zr[m] - mean) * rstd * g3[(size_t)s * 64 + m]
                       + be3[(size_t)s * 64 + m];
        o0 = fmaf(hn, W3[m * 2 + 0], o0);
        o1 = fmaf(hn, W3[m * 2 + 1], o1);
    }
    out[((size_t)b * 2 + 0) * S_ + s] = o0;   // [B, 2, S], coalesced over s
    out[((size_t)b * 2 + 1) * S_ + s] = o1;
}

// ---------------------------------------------------------------------------
extern "C" void kernel_launch(void* const* d_in, const int* in_sizes, int n_in,
                              void* d_out, int out_size, void* d_ws, size_t ws_size,
                              hipStream_t stream) {
    (void)in_sizes; (void)n_in; (void)out_size; (void)ws_size;

    const int*   inputs = (const int*)  d_in[0];
    const float* g1     = (const float*)d_in[1];
    const float* be1    = (const float*)d_in[2];
    const float* g2     = (const float*)d_in[3];
    const float* be2    = (const float*)d_in[4];
    const float* g3     = (const float*)d_in[5];
    const float* be3    = (const float*)d_in[6];
    const float* W1     = (const float*)d_in[7];
    const float* b1     = (const float*)d_in[8];
    const float* W2     = (const float*)d_in[9];
    const float* b2     = (const float*)d_in[10];
    const float* W3     = (const float*)d_in[11];
    const float* b3     = (const float*)d_in[12];
    float*       out    = (float*)d_out;

    // Workspace layout (floats)
    float* ws    = (float*)d_ws;
    float* acc   = ws;                                   // 256 (LN accumulators)
    float* base1 = acc + 256;                            // S*128
    float* hbuf  = base1 + (size_t)S_ * 128;             // B*S*128
    float* zbuf  = hbuf  + (size_t)B_ * S_ * 128;        // B*S*64

    zero_acc_kernel<<<1, 256, 0, stream>>>(acc);
    precompute_base1_kernel<<<S_ / 4, 128, 0, stream>>>(g1, be1, W1, b1, base1);
    stage1_kernel<<<B_ * S_, 128, 0, stream>>>(inputs, g1, W1, base1, hbuf, acc);

    const dim3 grid2(S_ / 32, B_);
    const size_t shmem2 = (size_t)(64 * 132 + 2 * 16 * 132 + 128) * sizeof(float);
    stage2_wmma_kernel<<<grid2, 64, shmem2, stream>>>(hbuf, g2, be2, W2, b2, zbuf, acc);

    const dim3 grid3(S_ / 128, B_);
    stage3_kernel<<<grid3, 128, 0, stream>>>(zbuf, g3, be3, W3, b3, acc, out);
}